// RGATConv_68092411510978
// MI455X (gfx1250) — compile-verified
//
#include <hip/hip_runtime.h>
#include <cstddef>

#define N_NODES 30000
#define N_EDGES 300000
#define N_REL   3
#define HEADS   4

typedef float v2f __attribute__((ext_vector_type(2)));
typedef float v8f __attribute__((ext_vector_type(8)));

__device__ __forceinline__ int enc_f(float f) {
    int b = __float_as_int(f);
    return b >= 0 ? b : (b ^ 0x7fffffff);
}
__device__ __forceinline__ float dec_f(int s) {
    return __int_as_float(s >= 0 ? s : (s ^ 0x7fffffff));
}

// ---------------------------------------------------------------------------
// Z[r] = A @ W[r]   (A: N x K row-major, W: R x K x COLS row-major)
// Block: 256 threads (8 wave32). Each block: 16 rows x COLS for one relation.
// FP32 WMMA 16x16x4. A chunk staged in LDS (padded, bank-conflict-free);
// B fragment reads are 64B-coalesced from L2-resident weights.
// ---------------------------------------------------------------------------
template <int K, int COLS>
__global__ __launch_bounds__(256) void gemm_wmma_f32(
    const float* __restrict__ A, const float* __restrict__ W,
    float* __restrict__ Z) {
    constexpr int TPW = COLS / 128;  // 16-col tiles per wave (8 waves)
    constexpr int KC  = 64;
    __shared__ float As[16 * 68];    // stride 68: banks (4*row + k) % 64 distinct

    const int r     = blockIdx.y;
    const int row0  = blockIdx.x * 16;
    const int tid   = threadIdx.x;
    const int lane  = tid & 31;
    const int wave  = tid >> 5;
    const int lrow  = lane & 15;          // A-row / C-col within tile
    const int khalf = (lane >> 4) * 2;    // this lane's K sub-offset
    const float* Wr = W + (size_t)r * K * COLS;

    v8f acc[TPW];
#pragma unroll
    for (int t = 0; t < TPW; ++t)
#pragma unroll
        for (int i = 0; i < 8; ++i) acc[t][i] = 0.0f;

    for (int kc = 0; kc < K; kc += KC) {
        // cooperative, coalesced stage of A[row0..row0+15][kc..kc+63]
#pragma unroll
        for (int i = 0; i < 4; ++i) {
            int e  = tid + i * 256;
            int ar = e >> 6, ac = e & 63;
            As[ar * 68 + ac] = A[(size_t)(row0 + ar) * K + kc + ac];
        }
        __syncthreads();
#pragma unroll
        for (int j = 0; j < KC / 4; ++j) {
            const int kk = j * 4 + khalf;
            v2f a;
            a.x = As[lrow * 68 + kk];
            a.y = As[lrow * 68 + kk + 1];
#pragma unroll
            for (int t = 0; t < TPW; ++t) {
                const int col   = (wave * TPW + t) * 16 + lrow;
                const float* bp = Wr + (size_t)(kc + kk) * COLS + col;
                v2f b;
                b.x = bp[0];
                b.y = bp[COLS];
                acc[t] = __builtin_amdgcn_wmma_f32_16x16x4_f32(
                    false, a, false, b, (short)0, acc[t], false, false);
            }
        }
        __syncthreads();
    }

    float* Zr       = Z + (size_t)r * N_NODES * COLS;
    const int rbase = row0 + (lane >> 4) * 8;
#pragma unroll
    for (int t = 0; t < TPW; ++t) {
        const int col = (wave * TPW + t) * 16 + lrow;
#pragma unroll
        for (int i = 0; i < 8; ++i)
            Zr[(size_t)(rbase + i) * COLS + col] = acc[t][i];
    }
}

// ---------------------------------------------------------------------------
// el[r,n,h] = <z[r,n,h,:], AL[r,h,:]>, er likewise. One thread per (r,n,h).
// ---------------------------------------------------------------------------
__global__ void attn_proj(const float* __restrict__ Z,
                          const float* __restrict__ AL,
                          const float* __restrict__ AR,
                          float* __restrict__ el, float* __restrict__ er,
                          int D) {
    int t = blockIdx.x * blockDim.x + threadIdx.x;
    if (t >= N_REL * N_NODES * HEADS) return;
    int h  = t & (HEADS - 1);
    int rn = t >> 2;  // r*N + n
    int r  = rn / N_NODES;
    const float* z  = Z + (size_t)rn * (HEADS * D) + h * D;
    const float* al = AL + ((size_t)r * HEADS + h) * D;
    const float* ar = AR + ((size_t)r * HEADS + h) * D;
    float sl = 0.f, sr = 0.f;
    for (int d = 0; d < D; ++d) {
        float zv = z[d];
        sl += zv * al[d];
        sr += zv * ar[d];
    }
    el[t] = sl;
    er[t] = sr;
}

__global__ void init_buffers(int* __restrict__ m, float* __restrict__ denom,
                             float* __restrict__ accum, int nmh, int nacc) {
    int t = blockIdx.x * blockDim.x + threadIdx.x;
    if (t < nmh) {
        m[t]     = (int)0x80000000;  // encodes -inf sentinel
        denom[t] = 0.f;
    }
    if (t < nacc) accum[t] = 0.f;
}

// pass 1: per-dst per-head segment max of leaky_relu(el[s]+er[d])
__global__ void edge_max(const int* __restrict__ src, const int* __restrict__ dst,
                         const float* __restrict__ el, const float* __restrict__ er,
                         int* __restrict__ m) {
    int t = blockIdx.x * blockDim.x + threadIdx.x;
    if (t >= N_REL * N_EDGES) return;
    int r = t / N_EDGES;
    int s = src[t], d = dst[t];
    const float* elp = el + ((size_t)r * N_NODES + s) * HEADS;
    const float* erp = er + ((size_t)r * N_NODES + d) * HEADS;
    int* mp = m + ((size_t)r * N_NODES + d) * HEADS;
#pragma unroll
    for (int h = 0; h < HEADS; ++h) {
        float v = elp[h] + erp[h];
        v = v > 0.f ? v : 0.2f * v;
        atomicMax(&mp[h], enc_f(v));
    }
}

// pass 2: w = exp(e - m[dst]); denom[dst] += w; stash w for the scatter pass
__global__ void edge_expw(const int* __restrict__ src, const int* __restrict__ dst,
                          const float* __restrict__ el, const float* __restrict__ er,
                          const int* __restrict__ m, float* __restrict__ denom,
                          float* __restrict__ wbuf) {
    int t = blockIdx.x * blockDim.x + threadIdx.x;
    if (t >= N_REL * N_EDGES) return;
    int r = t / N_EDGES;
    int s = src[t], d = dst[t];
    size_t nd = ((size_t)r * N_NODES + d) * HEADS;
    const float* elp = el + ((size_t)r * N_NODES + s) * HEADS;
    const float* erp = er + nd;
#pragma unroll
    for (int h = 0; h < HEADS; ++h) {
        float v = elp[h] + erp[h];
        v = v > 0.f ? v : 0.2f * v;
        float w = __expf(v - dec_f(m[nd + h]));
        wbuf[(size_t)t * HEADS + h] = w;
        atomicAdd(&denom[nd + h], w);
    }
}

// pass 3: accum[d] += (w/denom[d]) * z[s]; one thread per (edge, float4 chunk)
__global__ void edge_scatter(const int* __restrict__ src, const int* __restrict__ dst,
                             const float* __restrict__ wbuf,
                             const float* __restrict__ denom,
                             const float* __restrict__ Z,
                             float* __restrict__ accum, int D) {
    const int CH = HEADS * D / 4;
    long long t = (long long)blockIdx.x * blockDim.x + threadIdx.x;
    if (t >= (long long)N_REL * N_EDGES * CH) return;
    int c  = (int)(t % CH);
    int eg = (int)(t / CH);  // r*E + e
    int r  = eg / N_EDGES;
    int s  = src[eg], d = dst[eg];
    int h  = (c * 4) / D;
    float alpha = wbuf[(size_t)eg * HEADS + h] /
                  denom[((size_t)r * N_NODES + d) * HEADS + h];
    const float4 zv = *(const float4*)(Z + ((size_t)r * N_NODES + s) * (HEADS * D) + c * 4);
    float* op = accum + (size_t)d * (HEADS * D) + c * 4;
    atomicAdd(op + 0, alpha * zv.x);
    atomicAdd(op + 1, alpha * zv.y);
    atomicAdd(op + 2, alpha * zv.z);
    atomicAdd(op + 3, alpha * zv.w);
}

// h = relu(accum + sum_r b1[r])   (N x 128)
__global__ void finalize_relu(const float* __restrict__ accum,
                              const float* __restrict__ b, float* __restrict__ h) {
    int t = blockIdx.x * blockDim.x + threadIdx.x;
    if (t >= N_NODES * 128) return;
    int k   = t & 127;
    float v = accum[t] + b[k] + b[128 + k] + b[256 + k];
    h[t]    = v > 0.f ? v : 0.f;
}

// out[n,o] = mean_h( accum[n,h,o] + sum_r b2[r,h,o] )   (N x 64)
__global__ void finalize_mean(const float* __restrict__ accum,
                              const float* __restrict__ b, float* __restrict__ out) {
    int t = blockIdx.x * blockDim.x + threadIdx.x;
    if (t >= N_NODES * 64) return;
    int n = t >> 6, o = t & 63;
    float sum = 0.f;
#pragma unroll
    for (int h = 0; h < HEADS; ++h) {
        int k = h * 64 + o;
        sum += accum[(size_t)n * 256 + k] + b[k] + b[256 + k] + b[512 + k];
    }
    out[t] = sum * 0.25f;
}

static inline int cdiv(long long a, int b) { return (int)((a + b - 1) / b); }

extern "C" void kernel_launch(void* const* d_in, const int* in_sizes, int n_in,
                              void* d_out, int out_size, void* d_ws, size_t ws_size,
                              hipStream_t stream) {
    const float* x   = (const float*)d_in[0];
    const int*   src = (const int*)d_in[1];
    const int*   dst = (const int*)d_in[2];
    const float* W1  = (const float*)d_in[3];
    const float* al1 = (const float*)d_in[4];
    const float* ar1 = (const float*)d_in[5];
    const float* b1  = (const float*)d_in[6];
    const float* W2  = (const float*)d_in[7];
    const float* al2 = (const float*)d_in[8];
    const float* ar2 = (const float*)d_in[9];
    const float* b2  = (const float*)d_in[10];
    float* out = (float*)d_out;

    char*  ws  = (char*)d_ws;
    size_t off = 0;
    auto alloc = [&](size_t nfloats) {
        void* p = ws + off;
        off += ((nfloats * 4 + 255) / 256) * 256;
        return p;
    };
    float* Zbuf  = (float*)alloc((size_t)N_REL * N_NODES * 256);  // 92 MB (both layers)
    float* accum = (float*)alloc((size_t)N_NODES * 256);          // 31 MB (both layers)
    float* hbuf  = (float*)alloc((size_t)N_NODES * 128);
    float* el    = (float*)alloc((size_t)N_REL * N_NODES * HEADS);
    float* er    = (float*)alloc((size_t)N_REL * N_NODES * HEADS);
    int*   m     = (int*)alloc((size_t)N_REL * N_NODES * HEADS);
    float* denom = (float*)alloc((size_t)N_REL * N_NODES * HEADS);
    float* wbuf  = (float*)alloc((size_t)N_REL * N_EDGES * HEADS);

    const int nmh = N_REL * N_NODES * HEADS;
    const int tA  = N_REL * N_NODES * HEADS;
    const int tE  = N_REL * N_EDGES;
    dim3 gg(N_NODES / 16, N_REL);

    // ---------------- layer 1: x(256) -> 4 heads x 32 ----------------
    gemm_wmma_f32<256, 128><<<gg, 256, 0, stream>>>(x, W1, Zbuf);
    attn_proj<<<cdiv(tA, 256), 256, 0, stream>>>(Zbuf, al1, ar1, el, er, 32);
    {
        int nacc = N_NODES * 128;
        int imax = nmh > nacc ? nmh : nacc;
        init_buffers<<<cdiv(imax, 256), 256, 0, stream>>>(m, denom, accum, nmh, nacc);
    }
    edge_max<<<cdiv(tE, 256), 256, 0, stream>>>(src, dst, el, er, m);
    edge_expw<<<cdiv(tE, 256), 256, 0, stream>>>(src, dst, el, er, m, denom, wbuf);
    edge_scatter<<<cdiv((long long)tE * 32, 256), 256, 0, stream>>>(
        src, dst, wbuf, denom, Zbuf, accum, 32);
    finalize_relu<<<cdiv(N_NODES * 128, 256), 256, 0, stream>>>(accum, b1, hbuf);

    // ---------------- layer 2: h(128) -> 4 heads x 64 ----------------
    gemm_wmma_f32<128, 256><<<gg, 256, 0, stream>>>(hbuf, W2, Zbuf);
    attn_proj<<<cdiv(tA, 256), 256, 0, stream>>>(Zbuf, al2, ar2, el, er, 64);
    {
        int nacc = N_NODES * 256;
        int imax = nmh > nacc ? nmh : nacc;
        init_buffers<<<cdiv(imax, 256), 256, 0, stream>>>(m, denom, accum, nmh, nacc);
    }
    edge_max<<<cdiv(tE, 256), 256, 0, stream>>>(src, dst, el, er, m);
    edge_expw<<<cdiv(tE, 256), 256, 0, stream>>>(src, dst, el, er, m, denom, wbuf);
    edge_scatter<<<cdiv((long long)tE * 64, 256), 256, 0, stream>>>(
        src, dst, wbuf, denom, Zbuf, accum, 64);
    finalize_mean<<<cdiv(N_NODES * 64, 256), 256, 0, stream>>>(accum, b2, out);

    (void)in_sizes; (void)n_in; (void)out_size; (void)ws_size;
}